// TrajEmbedding_68255620268396
// MI455X (gfx1250) — compile-verified
//
#include <hip/hip_runtime.h>
#include <hip/hip_bf16.h>

#define N_NODES_C 100000
#define N_EDGES_C 1600000
#define FEAT_C 64
#define HEADS_C 8
#define OUTC_C 8
#define BATCH_C 64
#define MAXLEN_C 2048
#define NEG_SLOPE_C 0.2f
#define EMB_ELEMS (BATCH_C * MAXLEN_C * FEAT_C)

typedef float v2f __attribute__((ext_vector_type(2)));
typedef float v8f __attribute__((ext_vector_type(8)));

// ---------------------------------------------------------------------------
// Kernel 1: xw = x @ W  via V_WMMA_F32_16X16X4_F32 (exact fp32 matrix pipe).
// One wave computes a 16x64 output tile: 4 accumulators (16x16 each),
// 16 K-steps of 4. W (64x64) staged once per block in LDS; each wave stages
// its own 16x64 x-tile in LDS with coalesced float4 loads.
// ---------------------------------------------------------------------------
__global__ __launch_bounds__(256) void gat_gemm_xw(const float* __restrict__ x,
                                                   const float* __restrict__ W,
                                                   float* __restrict__ xw,
                                                   int n_tiles) {
    __shared__ float sW[64 * 64];       // 16 KB, row = k, col = n
    __shared__ float sX[8][16 * 64];    // 4 KB per wave

    const int tid  = threadIdx.x;
    const int wave = tid >> 5;
    const int lane = tid & 31;

    // Stage W: 1024 float4, 256 threads -> 4 each.
    {
        const float4* wg = (const float4*)W;
        float4* ws = (float4*)sW;
        #pragma unroll
        for (int i = 0; i < 4; ++i) ws[tid + i * 256] = wg[tid + i * 256];
    }

    const int tile   = blockIdx.x * 8 + wave;
    const int tile_c = tile < n_tiles ? tile : (n_tiles - 1);
    const int row0   = tile_c * 16;

    // Stage this wave's 16x64 x tile: 256 float4, 32 lanes -> 8 each.
    {
        const float4* xg = (const float4*)(x + (size_t)row0 * FEAT_C);
        float4* xs = (float4*)sX[wave];
        #pragma unroll
        for (int i = 0; i < 8; ++i) xs[i * 32 + lane] = xg[i * 32 + lane];
    }
    __syncthreads();

    if (tile < n_tiles) {
        const int m     = lane & 15;   // A-matrix: lanes 0-15 / 16-31 both map M=0..15
        const int n     = lane & 15;   // B/C: lane (mod 16) = column
        const int khalf = lane >> 4;   // selects K pair (A) / K row (B)
        const int chalf = lane >> 4;   // C: lanes 16-31 hold rows M+8

        v8f acc0 = {}, acc1 = {}, acc2 = {}, acc3 = {};
        const float* xt = sX[wave];

        #pragma unroll
        for (int k0 = 0; k0 < FEAT_C; k0 += 4) {
            v2f a;
            a.x = xt[m * FEAT_C + k0 + 2 * khalf + 0];
            a.y = xt[m * FEAT_C + k0 + 2 * khalf + 1];

            v2f b0, b1, b2, b3;
            const int kr0 = (k0 + 2 * khalf + 0) * 64;
            const int kr1 = (k0 + 2 * khalf + 1) * 64;
            b0.x = sW[kr0 + 0  + n];  b0.y = sW[kr1 + 0  + n];
            b1.x = sW[kr0 + 16 + n];  b1.y = sW[kr1 + 16 + n];
            b2.x = sW[kr0 + 32 + n];  b2.y = sW[kr1 + 32 + n];
            b3.x = sW[kr0 + 48 + n];  b3.y = sW[kr1 + 48 + n];

            acc0 = __builtin_amdgcn_wmma_f32_16x16x4_f32(false, a, false, b0, (short)0, acc0, false, false);
            acc1 = __builtin_amdgcn_wmma_f32_16x16x4_f32(false, a, false, b1, (short)0, acc1, false, false);
            acc2 = __builtin_amdgcn_wmma_f32_16x16x4_f32(false, a, false, b2, (short)0, acc2, false, false);
            acc3 = __builtin_amdgcn_wmma_f32_16x16x4_f32(false, a, false, b3, (short)0, acc3, false, false);
        }

        // D layout: VGPR r holds M=r (lanes 0-15) and M=r+8 (lanes 16-31), N=lane&15.
        float* outp = xw + (size_t)row0 * FEAT_C;
        #pragma unroll
        for (int r = 0; r < 8; ++r) {
            const int mr = r + 8 * chalf;
            outp[mr * FEAT_C + 0  + n] = acc0[r];
            outp[mr * FEAT_C + 16 + n] = acc1[r];
            outp[mr * FEAT_C + 32 + n] = acc2[r];
            outp[mr * FEAT_C + 48 + n] = acc3[r];
        }
    }
}

// ---------------------------------------------------------------------------
// Kernel 2: per (node, head): a_src/a_dst dots; init m=-inf, denom=0, out=0.
// ---------------------------------------------------------------------------
__global__ __launch_bounds__(256) void gat_prep(const float* __restrict__ xw,
                                                const float* __restrict__ att_src,
                                                const float* __restrict__ att_dst,
                                                float* __restrict__ a_src,
                                                float* __restrict__ a_dst,
                                                float* __restrict__ mmax,
                                                float* __restrict__ denom,
                                                float* __restrict__ outacc) {
    const int idx = blockIdx.x * blockDim.x + threadIdx.x;  // node*H + h
    if (idx >= N_NODES_C * HEADS_C) return;
    const int h = idx & (HEADS_C - 1);

    const float4* p  = (const float4*)(xw + (size_t)idx * OUTC_C);
    const float4* as = (const float4*)(att_src + h * OUTC_C);
    const float4* ad = (const float4*)(att_dst + h * OUTC_C);
    float4 x0 = p[0], x1 = p[1];
    float4 s0 = as[0], s1 = as[1];
    float4 d0 = ad[0], d1 = ad[1];

    a_src[idx] = x0.x * s0.x + x0.y * s0.y + x0.z * s0.z + x0.w * s0.w +
                 x1.x * s1.x + x1.y * s1.y + x1.z * s1.z + x1.w * s1.w;
    a_dst[idx] = x0.x * d0.x + x0.y * d0.y + x0.z * d0.z + x0.w * d0.w +
                 x1.x * d1.x + x1.y * d1.y + x1.z * d1.z + x1.w * d1.w;

    mmax[idx]  = __int_as_float(0xff800000);  // -inf
    denom[idx] = 0.0f;
    float4 z = {0.f, 0.f, 0.f, 0.f};
    ((float4*)(outacc + (size_t)idx * OUTC_C))[0] = z;
    ((float4*)(outacc + (size_t)idx * OUTC_C))[1] = z;
}

__device__ __forceinline__ void atomicMaxF(float* addr, float v) {
    // Ordered-int trick: signed max for v>=0, unsigned min for v<0.
    if (v >= 0.0f) atomicMax((int*)addr, __float_as_int(v));
    else           atomicMin((unsigned int*)addr, __float_as_uint(v));
}

__device__ __forceinline__ void load_edge(const long long* __restrict__ ei, int e,
                                          long long& s, long long& d) {
    if (e < N_EDGES_C) { s = ei[e]; d = ei[N_EDGES_C + e]; }
    else               { s = d = (long long)(e - N_EDGES_C); }  // self loop
}

// ---------------------------------------------------------------------------
// Kernel 3: segment max of leaky_relu(a_src[src] + a_dst[dst]) over dst.
// ---------------------------------------------------------------------------
__global__ __launch_bounds__(256) void gat_edge_max(const long long* __restrict__ ei,
                                                    const float* __restrict__ a_src,
                                                    const float* __restrict__ a_dst,
                                                    float* __restrict__ mmax) {
    const int e = blockIdx.x * blockDim.x + threadIdx.x;
    if (e >= N_EDGES_C + N_NODES_C) return;
    long long s, d;
    load_edge(ei, e, s, d);
    const float* ps = a_src + s * HEADS_C;
    const float* pd = a_dst + d * HEADS_C;
    float* pm = mmax + d * HEADS_C;
    #pragma unroll
    for (int h = 0; h < HEADS_C; ++h) {
        float v = ps[h] + pd[h];
        v = (v >= 0.0f) ? v : NEG_SLOPE_C * v;
        atomicMaxF(&pm[h], v);
    }
}

// ---------------------------------------------------------------------------
// Kernel 4: denom[dst] += exp(e - m[dst])   (native GLOBAL_ATOMIC_ADD_F32)
// ---------------------------------------------------------------------------
__global__ __launch_bounds__(256) void gat_edge_sum(const long long* __restrict__ ei,
                                                    const float* __restrict__ a_src,
                                                    const float* __restrict__ a_dst,
                                                    const float* __restrict__ mmax,
                                                    float* __restrict__ denom) {
    const int e = blockIdx.x * blockDim.x + threadIdx.x;
    if (e >= N_EDGES_C + N_NODES_C) return;
    long long s, d;
    load_edge(ei, e, s, d);
    const float* ps = a_src + s * HEADS_C;
    const float* pd = a_dst + d * HEADS_C;
    const float* pm = mmax + d * HEADS_C;
    float* pq = denom + d * HEADS_C;
    #pragma unroll
    for (int h = 0; h < HEADS_C; ++h) {
        float v = ps[h] + pd[h];
        v = (v >= 0.0f) ? v : NEG_SLOPE_C * v;
        atomicAdd(&pq[h], __expf(v - pm[h]));
    }
}

// ---------------------------------------------------------------------------
// Kernel 5: out[dst] += alpha * xw[src]; one thread per (edge, head).
// ---------------------------------------------------------------------------
__global__ __launch_bounds__(256) void gat_edge_scatter(const long long* __restrict__ ei,
                                                        const float* __restrict__ a_src,
                                                        const float* __restrict__ a_dst,
                                                        const float* __restrict__ mmax,
                                                        const float* __restrict__ denom,
                                                        const float* __restrict__ xw,
                                                        float* __restrict__ outacc) {
    const int idx = blockIdx.x * blockDim.x + threadIdx.x;
    if (idx >= (N_EDGES_C + N_NODES_C) * HEADS_C) return;
    const int e = idx >> 3;
    const int h = idx & 7;
    long long s, d;
    load_edge(ei, e, s, d);

    float v = a_src[s * HEADS_C + h] + a_dst[d * HEADS_C + h];
    v = (v >= 0.0f) ? v : NEG_SLOPE_C * v;
    const float alpha = __expf(v - mmax[d * HEADS_C + h]) / denom[d * HEADS_C + h];

    const float4* px = (const float4*)(xw + s * FEAT_C + h * OUTC_C);
    float4 x0 = px[0], x1 = px[1];
    float* po = outacc + d * FEAT_C + h * OUTC_C;
    atomicAdd(po + 0, alpha * x0.x);
    atomicAdd(po + 1, alpha * x0.y);
    atomicAdd(po + 2, alpha * x0.z);
    atomicAdd(po + 3, alpha * x0.w);
    atomicAdd(po + 4, alpha * x1.x);
    atomicAdd(po + 5, alpha * x1.y);
    atomicAdd(po + 6, alpha * x1.z);
    atomicAdd(po + 7, alpha * x1.w);
}

// ---------------------------------------------------------------------------
// Kernel 6: node_emb = relu(out + bias), in place.
// ---------------------------------------------------------------------------
__global__ __launch_bounds__(256) void gat_finalize(float* __restrict__ outacc,
                                                    const float* __restrict__ bias) {
    const int idx = blockIdx.x * blockDim.x + threadIdx.x;
    if (idx >= N_NODES_C * FEAT_C) return;
    const float v = outacc[idx] + bias[idx & (FEAT_C - 1)];
    outacc[idx] = v > 0.0f ? v : 0.0f;
}

// ---------------------------------------------------------------------------
// Kernel 7: padded gather emb[b,l,:] = node_emb[traj[b,l],:] * (l < len[b]),
// one thread per float4; first 64 threads also emit seq_lengths tail.
// ---------------------------------------------------------------------------
__global__ __launch_bounds__(256) void gat_gather(const float* __restrict__ node_emb,
                                                  const long long* __restrict__ traj,
                                                  const long long* __restrict__ seqlen,
                                                  float* __restrict__ out,
                                                  int out_size) {
    const int idx = blockIdx.x * blockDim.x + threadIdx.x;  // over B*L*16 float4
    const int total = BATCH_C * MAXLEN_C * (FEAT_C / 4);
    if (idx < total) {
        const int c4 = idx & 15;
        const int bl = idx >> 4;
        const int l  = bl & (MAXLEN_C - 1);
        const int b  = bl >> 11;
        float4 v = {0.f, 0.f, 0.f, 0.f};
        if ((long long)l < seqlen[b]) {
            const long long node = traj[b * MAXLEN_C + l];
            v = ((const float4*)(node_emb + node * FEAT_C))[c4];
        }
        ((float4*)out)[bl * 16 + c4] = v;
    }
    if (idx < BATCH_C && out_size >= EMB_ELEMS + 2 * BATCH_C) {
        ((long long*)out)[EMB_ELEMS / 2 + idx] = seqlen[idx];  // int64 tail
    }
}

// ---------------------------------------------------------------------------
extern "C" void kernel_launch(void* const* d_in, const int* in_sizes, int n_in,
                              void* d_out, int out_size, void* d_ws, size_t ws_size,
                              hipStream_t stream) {
    const float* x        = (const float*)d_in[0];
    const float* W        = (const float*)d_in[1];
    const float* att_src  = (const float*)d_in[2];
    const float* att_dst  = (const float*)d_in[3];
    const float* bias     = (const float*)d_in[4];
    const long long* ei   = (const long long*)d_in[5];
    const long long* traj = (const long long*)d_in[6];
    const long long* slen = (const long long*)d_in[7];
    float* out = (float*)d_out;

    // Workspace layout (floats).
    float* ws     = (float*)d_ws;
    float* xw     = ws;                                  // N*64
    float* a_src  = xw + (size_t)N_NODES_C * FEAT_C;     // N*8
    float* a_dst  = a_src + (size_t)N_NODES_C * HEADS_C; // N*8
    float* mmax   = a_dst + (size_t)N_NODES_C * HEADS_C; // N*8
    float* denom  = mmax + (size_t)N_NODES_C * HEADS_C;  // N*8
    float* outacc = denom + (size_t)N_NODES_C * HEADS_C; // N*64

    const int n_tiles = N_NODES_C / 16;                  // 6250, exact
    const int gemm_blocks = (n_tiles + 7) / 8;
    gat_gemm_xw<<<gemm_blocks, 256, 0, stream>>>(x, W, xw, n_tiles);

    const int nh = N_NODES_C * HEADS_C;
    gat_prep<<<(nh + 255) / 256, 256, 0, stream>>>(xw, att_src, att_dst,
                                                   a_src, a_dst, mmax, denom, outacc);

    const int etot = N_EDGES_C + N_NODES_C;
    gat_edge_max<<<(etot + 255) / 256, 256, 0, stream>>>(ei, a_src, a_dst, mmax);
    gat_edge_sum<<<(etot + 255) / 256, 256, 0, stream>>>(ei, a_src, a_dst, mmax, denom);

    const int eh = etot * HEADS_C;
    gat_edge_scatter<<<(eh + 255) / 256, 256, 0, stream>>>(ei, a_src, a_dst, mmax,
                                                           denom, xw, outacc);

    const int nf = N_NODES_C * FEAT_C;
    gat_finalize<<<(nf + 255) / 256, 256, 0, stream>>>(outacc, bias);

    const int gth = BATCH_C * MAXLEN_C * (FEAT_C / 4);
    gat_gather<<<(gth + 255) / 256, 256, 0, stream>>>(outacc, traj, slen, out, out_size);
}